// Graph_Pool_27736898798369
// MI455X (gfx1250) — compile-verified
//
#include <hip/hip_runtime.h>
#include <math.h>

typedef float v2f __attribute__((ext_vector_type(2)));
typedef float v8f __attribute__((ext_vector_type(8)));
typedef float f4  __attribute__((ext_vector_type(4)));

#define NN   16384
#define DD   256
#define KSEL 8192

// ---------------------------------------------------------------------------
// Kernel 1: scores = sigmoid(x @ W + b) via V_WMMA_F32_16X16X4_F32 (full f32).
// One wave (32 lanes) computes a 16-row tile. A-matrix 16x4 f32 layout:
// lanes 0-15 hold K=0,1 (VGPR0,1), lanes 16-31 hold K=2,3. B = W broadcast
// across 16 (identical) columns with the mirrored K split.
// ---------------------------------------------------------------------------
__global__ __launch_bounds__(32) void scores_wmma(const float* __restrict__ x,
                                                  const float* __restrict__ W,
                                                  const float* __restrict__ b,
                                                  float* __restrict__ scores) {
  const int lane = threadIdx.x;            // 0..31, EXEC all ones around the WMMAs
  const int half = lane >> 4;              // 0 or 1 -> which K pair this lane holds
  const int m0   = blockIdx.x * 16;
  const int mrow = m0 + (lane & 15);
  const float* xr = x + (size_t)mrow * DD + (half << 1);

  v8f c = {};
  #pragma unroll 8
  for (int k = 0; k < DD; k += 4) {
    v2f a = *(const v2f*)(xr + k);         // x[mrow][k + 2*half .. +1]
    v2f bb;
    bb.x = W[k + (half << 1)];             // B row K=2*half   (all 16 cols equal)
    bb.y = W[k + (half << 1) + 1];         // B row K=2*half+1
    c = __builtin_amdgcn_wmma_f32_16x16x4_f32(
        /*neg_a=*/false, a, /*neg_b=*/false, bb,
        /*c_mod=*/(short)0, c, /*reuse_a=*/false, /*reuse_b=*/false);
  }

  // D layout: VGPR v, lanes 0-15 -> M=v, lanes 16-31 -> M=8+v (all N identical).
  const float bias = b[0];
  if ((lane & 15) == 0) {
    const int base = m0 + half * 8;
    #pragma unroll
    for (int v = 0; v < 8; ++v) {
      float z = c[v] + bias;
      scores[base + v] = 1.0f / (1.0f + expf(-z));
    }
  }
}

// ---------------------------------------------------------------------------
// Kernel 2: exact top-k with ordering identical to jax.lax.top_k.
// Full bitonic sort of 16384 64-bit keys in 128 KB LDS (fits CDNA5 320KB WGP).
// key = (0xFFFFFFFF - bits(score)) << 32 | index  -> ascending sort gives
// score-descending, tie -> smaller index first. Scores are sigmoid outputs
// (positive), so the float bit pattern is monotone in value.
// ---------------------------------------------------------------------------
__global__ __launch_bounds__(1024) void topk_sort(const float* __restrict__ scores,
                                                  int* __restrict__ idx_out,
                                                  float* __restrict__ val_out,
                                                  float* __restrict__ idx_f_out) {
  extern __shared__ unsigned long long keys[];   // 16384 * 8B = 128 KB dynamic LDS
  const int t = threadIdx.x;

  for (int i = t; i < NN; i += 1024) {
    unsigned sb = __float_as_uint(scores[i]);
    keys[i] = ((unsigned long long)(0xFFFFFFFFu - sb) << 32) | (unsigned)i;
  }
  __syncthreads();

  for (int kk = 2; kk <= NN; kk <<= 1) {
    for (int j = kk >> 1; j > 0; j >>= 1) {
      for (int i = t; i < NN; i += 1024) {
        int ixj = i ^ j;
        if (ixj > i) {
          unsigned long long a = keys[i], bb = keys[ixj];
          bool asc = ((i & kk) == 0);
          if (asc ? (a > bb) : (a < bb)) { keys[i] = bb; keys[ixj] = a; }
        }
      }
      __syncthreads();
    }
  }

  for (int i = t; i < KSEL; i += 1024) {
    unsigned long long key = keys[i];
    int   idx = (int)(unsigned)(key & 0xFFFFFFFFull);
    float val = __uint_as_float(0xFFFFFFFFu - (unsigned)(key >> 32));
    idx_out[i]   = idx;
    val_out[i]   = val;
    idx_f_out[i] = (float)idx;               // third reference output, value-cast
  }
}

// ---------------------------------------------------------------------------
// Kernel 3: x_pooled[i] = x[idx[i]] * vals[i]   (one block per row, float4)
// ---------------------------------------------------------------------------
__global__ __launch_bounds__(64) void gather_x(const float* __restrict__ x,
                                               const int* __restrict__ idx,
                                               const float* __restrict__ vals,
                                               float* __restrict__ outx) {
  const int i  = blockIdx.x;
  const int ri = idx[i];
  const float v = vals[i];
  const f4* src = (const f4*)(x + (size_t)ri * DD);
  f4*       dst = (f4*)(outx + (size_t)i * DD);
  f4 a = src[threadIdx.x];
  a.x *= v; a.y *= v; a.z *= v; a.w *= v;
  dst[threadIdx.x] = a;
}

// ---------------------------------------------------------------------------
// Kernel 4 (bandwidth-dominant): A_pooled[i][j] = A[idx[i]][idx[j]]^2.
// Stage the full gathered 64 KB row directly into LDS with the CDNA5 async
// DMA path (GLOBAL_LOAD_ASYNC_TO_LDS_B128, ASYNCcnt-tracked): no VGPR
// round-trip, no ds_store port traffic, more issue slots free to overlap the
// previous row's gather. Then ds_load gathers + coalesced NT b32 stores.
// ~768 MB total traffic -> ~33 us HBM floor at 23.3 TB/s.
// ---------------------------------------------------------------------------
__global__ __launch_bounds__(256) void gather_A(const float* __restrict__ A,
                                                const int* __restrict__ idx,
                                                float* __restrict__ outA) {
  extern __shared__ float row[];               // 16384 floats = 64 KB dynamic LDS
  const int i = blockIdx.x;
  const int t = threadIdx.x;
  const int ri = idx[i];

  const f4* src4 = (const f4*)(A + (size_t)ri * NN);
  // Flat->LDS aperture mapping: low 32 bits of the generic shared pointer are
  // the DS byte offset the async DMA adds to LDS_BASE.
  const unsigned ldsbase = (unsigned)(unsigned long long)(const void*)row;

  #pragma unroll 4
  for (int p = t; p < NN / 4; p += 256) {
    unsigned ldsa = ldsbase + (unsigned)(p * 16);
    unsigned long long gaddr = (unsigned long long)(const void*)(src4 + p);
    // ISA 15.18.3 opcode 98: copies 16B per lane from global memory to LDS at vdst.
    asm volatile("global_load_async_to_lds_b128 %0, %1, off"
                 :: "v"(ldsa), "v"(gaddr) : "memory");
  }
  asm volatile("s_wait_asynccnt 0" ::: "memory");   // this wave's DMA done
  __syncthreads();                                   // all waves' rows visible

  float* outr = outA + (size_t)i * KSEL;
  #pragma unroll 4
  for (int j = t; j < KSEL; j += 256) {
    float v = row[idx[j]];                     // LDS gather; idx[] stays hot in WGP$
    __builtin_nontemporal_store(v * v, outr + j);
  }
}

// ---------------------------------------------------------------------------
extern "C" void kernel_launch(void* const* d_in, const int* in_sizes, int n_in,
                              void* d_out, int out_size, void* d_ws, size_t ws_size,
                              hipStream_t stream) {
  (void)in_sizes; (void)n_in; (void)out_size; (void)ws_size;

  const float* A = (const float*)d_in[0];   // (N, N)
  const float* x = (const float*)d_in[1];   // (N, D)
  const float* W = (const float*)d_in[2];   // (D, 1)
  const float* b = (const float*)d_in[3];   // (1,)
  // d_in[4] = k (always 8192 for this problem instance)

  float* outA   = (float*)d_out;                       // 8192*8192
  float* outx   = outA + (size_t)KSEL * KSEL;          // 8192*256
  float* outidx = outx + (size_t)KSEL * DD;            // 8192

  float* scores = (float*)d_ws;                        // 16384 f32
  int*   idx    = (int*)(scores + NN);                 // 8192 i32
  float* vals   = (float*)(idx + KSEL);                // 8192 f32

  scores_wmma<<<NN / 16, 32, 0, stream>>>(x, W, b, scores);
  topk_sort<<<1, 1024, NN * sizeof(unsigned long long), stream>>>(scores, idx, vals, outidx);
  gather_x<<<KSEL, 64, 0, stream>>>(x, idx, vals, outx);
  gather_A<<<KSEL, 256, NN * sizeof(float), stream>>>(A, idx, outA);
}